// TOM_Initialization_4664334483660
// MI455X (gfx1250) — compile-verified
//
#include <hip/hip_runtime.h>

typedef __attribute__((ext_vector_type(2))) float v2f;
typedef __attribute__((ext_vector_type(4))) float v4f;
typedef __attribute__((ext_vector_type(8))) float v8f;

#define B_ROWS 8192
#define N_OUT  2048
#define K_DIM  8192

#define BM 128
#define BN 128
#define BK 16
#define LDP 18   // LDS row pitch in floats (pad vs 64 banks, keeps 8B alignment)

__global__ __launch_bounds__(256)
void gemm_perm_f32_wmma(const float* __restrict__ weight,
                        const float* __restrict__ fc_w,
                        const float* __restrict__ fc_b,
                        const int*   __restrict__ swap1,
                        const int*   __restrict__ swap2,
                        float* __restrict__ out)
{
    __shared__ float As[BM * LDP];     // A tile: [row 0..127][k 0..15]
    __shared__ float Bs[BN * LDP];     // B tile: [n   0..127][k 0..15] (fc_w layout)
    __shared__ int   roff[BM][4];      // per-row remapped chunk base offsets (elements)

    const int tid  = threadIdx.x;
    const int lane = tid & 31;
    const int wave = tid >> 5;
    const int wm   = wave & 3;         // 4 waves along M (32 rows each)
    const int wn   = wave >> 2;        // 2 waves along N (64 cols each)

    const int m0 = blockIdx.y * BM;
    const int n0 = blockIdx.x * BN;

    // Per-row K-chunk permutation: sequential swaps (1,2) then (2,3).
    if (tid < BM) {
        const int g  = m0 + tid;
        const int s1 = swap1[g];
        const int s2 = swap2[g];
        const int p1  = s1 ? 2 : 1;
        const int mid = s1 ? 1 : 2;
        const int p2  = s2 ? 3 : mid;
        const int p3  = s2 ? mid : 3;
        roff[tid][0] = 0;
        roff[tid][1] = p1 << 11;       // * 2048 elements
        roff[tid][2] = p2 << 11;
        roff[tid][3] = p3 << 11;
    }
    __syncthreads();

    // Each thread stages the same two rows every iteration: hoist the
    // remapped chunk offsets for those rows into registers.
    const int r0 = tid >> 2;           // 0..63
    const int r1 = r0 + 64;            // 64..127
    const int c  = (tid & 3) << 2;     // 0,4,8,12

    int offA0[4], offA1[4];
#pragma unroll
    for (int j = 0; j < 4; ++j) {
        offA0[j] = roff[r0][j];
        offA1[j] = roff[r1][j];
    }

    const float* aBase0 = weight + (size_t)(m0 + r0) * K_DIM + c;
    const float* aBase1 = weight + (size_t)(m0 + r1) * K_DIM + c;
    const float* bBase0 = fc_w + (size_t)(n0 + r0) * K_DIM + c;
    const float* bBase1 = fc_w + (size_t)(n0 + r1) * K_DIM + c;

    v8f acc[2][4];
#pragma unroll
    for (int mt = 0; mt < 2; ++mt)
#pragma unroll
        for (int nt = 0; nt < 4; ++nt)
            acc[mt][nt] = (v8f){0.f, 0.f, 0.f, 0.f, 0.f, 0.f, 0.f, 0.f};

    const int lrow  = lane & 15;       // row-in-tile for A/B fragments, N for C/D
    const int khalf = (lane >> 4) * 2; // K sub-offset per lane half

    // Prime the register buffer with tile k0 = 0.
    v4f aReg0 = *(const v4f*)(aBase0 + offA0[0]);
    v4f aReg1 = *(const v4f*)(aBase1 + offA1[0]);
    v4f bReg0 = *(const v4f*)(bBase0);
    v4f bReg1 = *(const v4f*)(bBase1);

    for (int k0 = 0; k0 < K_DIM; k0 += BK) {
        // Commit current register buffer to LDS.
        *(v4f*)&As[r0 * LDP + c] = aReg0;
        *(v4f*)&As[r1 * LDP + c] = aReg1;
        *(v4f*)&Bs[r0 * LDP + c] = bReg0;
        *(v4f*)&Bs[r1 * LDP + c] = bReg1;
        __syncthreads();

        // Prefetch next tile into registers; these loads stay in flight
        // across the whole WMMA block below.
        const int k0n = k0 + BK;
        if (k0n < K_DIM) {
            const int kch = k0n >> 11;     // BK=16 never straddles a 2048-chunk
            const int kin = k0n & 2047;
            aReg0 = *(const v4f*)(aBase0 + offA0[kch] + kin);
            aReg1 = *(const v4f*)(aBase1 + offA1[kch] + kin);
            bReg0 = *(const v4f*)(bBase0 + k0n);
            bReg1 = *(const v4f*)(bBase1 + k0n);
        }

#pragma unroll
        for (int kk = 0; kk < BK; kk += 4) {
            v2f afrag[2];
#pragma unroll
            for (int mt = 0; mt < 2; ++mt) {
                const int row = wm * 32 + mt * 16 + lrow;
                afrag[mt] = *(const v2f*)&As[row * LDP + kk + khalf];
            }
            v2f bfrag[4];
#pragma unroll
            for (int nt = 0; nt < 4; ++nt) {
                const int nrow = wn * 64 + nt * 16 + lrow;
                bfrag[nt] = *(const v2f*)&Bs[nrow * LDP + kk + khalf];
            }
#pragma unroll
            for (int mt = 0; mt < 2; ++mt)
#pragma unroll
                for (int nt = 0; nt < 4; ++nt)
                    acc[mt][nt] = __builtin_amdgcn_wmma_f32_16x16x4_f32(
                        /*neg_a=*/false, afrag[mt],
                        /*neg_b=*/false, bfrag[nt],
                        /*c_mod=*/(short)0, acc[mt][nt],
                        /*reuse_a=*/false, /*reuse_b=*/false);
        }
        __syncthreads();
    }

    // Epilogue: C/D layout => lanes 0-15: M=v, N=lane; lanes 16-31: M=v+8, N=lane-16.
#pragma unroll
    for (int mt = 0; mt < 2; ++mt) {
        const int rbase = m0 + wm * 32 + mt * 16 + ((lane >> 4) << 3);
#pragma unroll
        for (int nt = 0; nt < 4; ++nt) {
            const int col  = n0 + wn * 64 + nt * 16 + lrow;
            const float bv = fc_b[col];
#pragma unroll
            for (int v = 0; v < 8; ++v) {
                out[(size_t)(rbase + v) * N_OUT + col] = acc[mt][nt][v] + bv;
            }
        }
    }
}

extern "C" void kernel_launch(void* const* d_in, const int* in_sizes, int n_in,
                              void* d_out, int out_size, void* d_ws, size_t ws_size,
                              hipStream_t stream) {
    const float* weight = (const float*)d_in[0];   // [8192, 8192]
    const float* fc_w   = (const float*)d_in[1];   // [2048, 8192]
    const float* fc_b   = (const float*)d_in[2];   // [2048]
    const int*   swap1  = (const int*)d_in[3];     // [8192]
    const int*   swap2  = (const int*)d_in[4];     // [8192]
    float* out = (float*)d_out;                    // [8192, 2048]

    dim3 grid(N_OUT / BN, B_ROWS / BM);            // (16, 64)
    dim3 block(256);
    gemm_perm_f32_wmma<<<grid, block, 0, stream>>>(weight, fc_w, fc_b, swap1, swap2, out);
}